// Neuromorphizer_86457691668532
// MI455X (gfx1250) — compile-verified
//
#include <hip/hip_runtime.h>
#include <cstdint>
#include <cstddef>

// ---------------------------------------------------------------------------
// Neuromorphizer: temporal scan over (T,H,W) frames producing event frames.
// Memory-bound streaming kernel: ~0.73 GB moved -> ~31us at 23.3 TB/s.
// Strategy:
//   * 1 thread = 4 consecutive pixels (float4 / b128 traffic everywhere)
//   * st/ts recurrent state lives in VGPRs for all 96 steps
//   * 10+10 noise bytes/pixel pre-packed into two 10-bit register masks
//   * frame stream staged through LDS with gfx1250 async loads
//     (global_load_async_to_lds_b128, ASYNCcnt-tracked, double buffered)
//   * non-temporal b128 stores for the 354MB output stream (keep L2 for noise)
// ---------------------------------------------------------------------------

typedef float f4 __attribute__((ext_vector_type(4)));

#define T_FRAMES 96
#define H_ 720
#define W_ 1280
#define HW_ (H_ * W_)            // 921600, divisible by 1024
#define N_NOISE 10
#define DELTA_T_US (1000000.0f / 30.0f)
#define BLK 256                  // 8 wave32 per block
#define PIX_PER_BLK (BLK * 4)    // 1024 pixels -> 4KB frame tile per block

// Issue one 16-byte async global->LDS transfer for this lane.
// VDST operand = LDS byte offset (low 32 bits of the flat shared pointer),
// VADDR = 64-bit global address, GV addressing mode ("off").
__device__ __forceinline__ void async_load16(uint32_t lds_off, const void* gptr) {
    asm volatile("global_load_async_to_lds_b128 %0, %1, off"
                 :: "v"(lds_off), "v"((uint64_t)(uintptr_t)gptr)
                 : "memory");
}

__global__ __launch_bounds__(BLK) void neuromorph_scan_kernel(
    const float* __restrict__ tensor,      // (T,H,W) f32
    const float* __restrict__ state,       // (H,W)  f32
    const float* __restrict__ tsurf,       // (H,W)  f32
    const unsigned char* __restrict__ on_noise,   // (10,H,W) bool(u8)
    const unsigned char* __restrict__ off_noise,  // (10,H,W) bool(u8)
    float* __restrict__ out)               // (T,H,W) f32
{
    __shared__ __align__(16) f4 smem[2][BLK];   // double-buffered frame tile, 8KB

    const int      tid = threadIdx.x;
    const size_t   p   = ((size_t)blockIdx.x * BLK + (size_t)tid) * 4; // base pixel

    // Recurrent per-pixel state in registers (b128 loads, 16B aligned).
    f4 st = *(const f4*)(state + p);
    f4 ts = *(const f4*)(tsurf + p);

    // Pack the 10 on/off noise planes into per-pixel 10-bit masks.
    // 20 dword loads per thread, then zero noise traffic in the time loop.
    uint32_t onm[4]  = {0u, 0u, 0u, 0u};
    uint32_t offm[4] = {0u, 0u, 0u, 0u};
#pragma unroll
    for (int k = 0; k < N_NOISE; ++k) {
        uint32_t v = *(const uint32_t*)(on_noise  + (size_t)k * HW_ + p);
        uint32_t w = *(const uint32_t*)(off_noise + (size_t)k * HW_ + p);
#pragma unroll
        for (int j = 0; j < 4; ++j) {
            if ((v >> (8 * j)) & 0xFFu) onm[j]  |= (1u << k);
            if ((w >> (8 * j)) & 0xFFu) offm[j] |= (1u << k);
        }
    }

    // LDS byte offset of this lane's slot (low 32 bits of flat LDS pointer).
    const uint32_t lds0 = (uint32_t)(uintptr_t)(&smem[0][0]) + (uint32_t)tid * 16u;

    // Prime the pipeline with frame 0.
    async_load16(lds0, tensor + p);

    int idx = 1;  // (t_idx+1) % N_NOISE, maintained incrementally

#pragma unroll 2
    for (int t = 0; t < T_FRAMES; ++t) {
        if (t + 1 < T_FRAMES) {
            // Prefetch next frame tile into the other LDS buffer, then wait
            // until only that one transfer is outstanding (in-order ASYNCcnt
            // completion => frame t is resident).
            async_load16(lds0 + (uint32_t)(((t + 1) & 1) * (BLK * 16)),
                         tensor + (size_t)(t + 1) * HW_ + p);
            asm volatile("s_wait_asynccnt 1" ::: "memory");
        } else {
            asm volatile("s_wait_asynccnt 0" ::: "memory");
        }

        const f4 x = smem[t & 1][tid];     // ds_load_b128, own slot, no barrier

        const float fcnt     = (float)(t + 1);
        const float t_us     = fcnt * DELTA_T_US;
        const float min_time = __builtin_floorf(t_us / DELTA_T_US);

        f4 o;
#pragma unroll
        for (int j = 0; j < 4; ++j) {
            const float idle = (ts[j] <= min_time) ? 1.0f : 0.0f;
            const float diff = (x[j] - st[j]) * idle;
            const bool  zero = (diff == 0.0f);        // |diff| <= THRESHOLD(=0)
            float r = zero ? 127.0f : ((diff > 0.0f) ? 255.0f : 0.0f);
            if (!zero) { st[j] = x[j]; ts[j] = fcnt; }
            if ((onm[j]  >> idx) & 1u) r = 255.0f;    // on-noise
            if ((offm[j] >> idx) & 1u) r = 0.0f;      // off-noise wins
            o[j] = r;
        }

        // Streaming output: non-temporal b128 store.
        __builtin_nontemporal_store(o, (f4*)(out + (size_t)t * HW_ + p));

        idx = (idx == N_NOISE - 1) ? 0 : idx + 1;
    }
}

extern "C" void kernel_launch(void* const* d_in, const int* in_sizes, int n_in,
                              void* d_out, int out_size, void* d_ws, size_t ws_size,
                              hipStream_t stream) {
    (void)in_sizes; (void)n_in; (void)out_size; (void)d_ws; (void)ws_size;

    const float*         tensor   = (const float*)d_in[0];
    const float*         state    = (const float*)d_in[1];
    const float*         tsurf    = (const float*)d_in[2];
    const unsigned char* on_noise = (const unsigned char*)d_in[3];
    const unsigned char* off_noise= (const unsigned char*)d_in[4];
    float*               out      = (float*)d_out;

    const int blocks = HW_ / PIX_PER_BLK;   // 900, exact cover -> EXEC all-ones
    neuromorph_scan_kernel<<<blocks, BLK, 0, stream>>>(
        tensor, state, tsurf, on_noise, off_noise, out);
}